// RegLSTM_45148696215787
// MI455X (gfx1250) — compile-verified
//
#include <hip/hip_runtime.h>
#include <stdint.h>

#define S_LEN 4096
#define D_DIM 64
#define M_HID 128
#define G_DIM 512   // 4*M

typedef float v2f __attribute__((ext_vector_type(2)));
typedef float v8f __attribute__((ext_vector_type(8)));

__device__ __forceinline__ float fast_rcp(float x) { return __builtin_amdgcn_rcpf(x); }

// sigmoid via v_exp + v_rcp (no IEEE division on the recurrence critical path).
// exp(-x) -> inf for very negative x: rcp(inf) = 0, which is the correct limit.
__device__ __forceinline__ float sigmf(float x) {
    return fast_rcp(1.0f + __expf(-x));
}
// tanh(x) = (e-1)/(e+1), e = exp(2x) clamped so e stays finite (no inf*0 = NaN).
__device__ __forceinline__ float tanh_fast(float x) {
    float e = __expf(fminf(fmaxf(2.0f * x, -30.0f), 30.0f));
    return (e - 1.0f) * fast_rcp(e + 1.0f);
}

__device__ __forceinline__ uint32_t f2bf(float f) {
    uint32_t b = __float_as_uint(f);
    b += 0x7fffu + ((b >> 16) & 1u);   // round-to-nearest-even
    return b >> 16;
}
__device__ __forceinline__ uint32_t pack2bf(float a, float b) {
    return f2bf(a) | (f2bf(b) << 16);
}
__device__ __forceinline__ float bf_lo(uint32_t u) { return __uint_as_float(u << 16); }
__device__ __forceinline__ float bf_hi(uint32_t u) { return __uint_as_float(u & 0xffff0000u); }

// ---------------------------------------------------------------------------
// out[s,e] = act( sum_d X[s,d] * W[e,d] + bias[e] )   (4096x64 @ 64x64^T)
// One 16x16 output tile per wave via V_WMMA_F32_16X16X4_F32, K-loop of 16.
// act: 0 = relu, 1 = sigmoid
// ---------------------------------------------------------------------------
__global__ void proj64_wmma(const float* __restrict__ X, const float* __restrict__ W,
                            const float* __restrict__ bias, float* __restrict__ out,
                            int act)
{
    const int lane = threadIdx.x & 31;
    const int wave = threadIdx.x >> 5;
    const int tile = blockIdx.x * 8 + wave;       // 1024 tiles total
    const int tm = tile >> 2;                     // S/16 tile
    const int tn = tile & 3;                      // D/16 tile
    const int r  = lane & 15;
    const int kb = (lane >> 4) << 1;              // 0 (lanes 0-15) or 2 (lanes 16-31)

    const float* xrow = X + (size_t)(tm * 16 + r) * D_DIM;  // A row (M=r)
    const float* wrow = W + (size_t)(tn * 16 + r) * D_DIM;  // B col (N=r): B[k][n]=W[n][k]

    v8f acc = {};
#pragma unroll
    for (int kk = 0; kk < 16; ++kk) {
        const int k0 = kk * 4;
        v2f a, b;
        // 32-bit A 16x4 layout: lane<16 -> K=0,1 ; lane>=16 -> K=2,3
        a.x = xrow[k0 + kb];  a.y = xrow[k0 + kb + 1];
        // 32-bit B 4x16 layout (rows striped across VGPRs, cols across lanes)
        b.x = wrow[k0 + kb];  b.y = wrow[k0 + kb + 1];
        acc = __builtin_amdgcn_wmma_f32_16x16x4_f32(false, a, false, b,
                                                    (short)0, acc, false, false);
    }
    // C/D layout: VGPR v -> row v (lanes 0-15) / row v+8 (lanes 16-31), col = lane&15
    const int col = tn * 16 + r;
    const float bv = bias[col];
    const int rbase = tm * 16 + ((lane >> 4) << 3);
#pragma unroll
    for (int v = 0; v < 8; ++v) {
        float val = acc[v] + bv;
        val = act ? sigmf(val) : fmaxf(val, 0.0f);
        out[(size_t)(rbase + v) * D_DIM + col] = val;
    }
}

// ---------------------------------------------------------------------------
// Layer-0 grouped LSTM: 1 workgroup per channel, 512 threads (1 per gate row).
// Whh0[c] (512x128 f32 = 256KB) held in LDS for all 4096 steps, transposed to
// [m/4][j][4] so per-m4 float4 loads are bank-conflict-free across j.
// ---------------------------------------------------------------------------
__global__ void lstm0_persistent(const float* __restrict__ Wih, const float* __restrict__ Whh,
                                 const float* __restrict__ bih, const float* __restrict__ bhh,
                                 const float* __restrict__ u, uint16_t* __restrict__ h1)
{
    extern __shared__ char smem[];
    float4* Wt4  = (float4*)smem;                              // [32][512] float4
    float*  hbuf = (float*)(smem + (size_t)G_DIM * M_HID * 4); // [128]
    float*  gbuf = hbuf + M_HID;                               // [512]

    const int c = blockIdx.x;
    const int j = threadIdx.x;    // 0..511 (gate index)

    const float4* gw = (const float4*)(Whh + ((size_t)c * G_DIM + j) * M_HID);
#pragma unroll
    for (int m4 = 0; m4 < 32; ++m4)
        Wt4[m4 * G_DIM + j] = gw[m4];

    const float wih_j = Wih[(size_t)c * G_DIM + j];
    const float bsum  = bih[(size_t)c * G_DIM + j] + bhh[(size_t)c * G_DIM + j];

    float creg = 0.0f;
    if (j < M_HID) hbuf[j] = 0.0f;
    __syncthreads();

    const float4* h4 = (const float4*)hbuf;
    for (int t = 0; t < S_LEN; ++t) {
        const float ut = u[t * D_DIM + c];          // workgroup-uniform load
        // dual accumulators: break the 128-long serial FMA chain into 2x64
        float acc0 = fmaf(wih_j, ut, bsum);
        float acc1 = 0.0f;
#pragma unroll
        for (int m4 = 0; m4 < 32; m4 += 2) {
            float4 w0 = Wt4[(m4 + 0) * G_DIM + j];
            float4 h0 = h4[m4 + 0];                 // LDS broadcast
            float4 w1 = Wt4[(m4 + 1) * G_DIM + j];
            float4 h1v = h4[m4 + 1];
            acc0 = fmaf(w0.x, h0.x, acc0);
            acc1 = fmaf(w1.x, h1v.x, acc1);
            acc0 = fmaf(w0.y, h0.y, acc0);
            acc1 = fmaf(w1.y, h1v.y, acc1);
            acc0 = fmaf(w0.z, h0.z, acc0);
            acc1 = fmaf(w1.z, h1v.z, acc1);
            acc0 = fmaf(w0.w, h0.w, acc0);
            acc1 = fmaf(w1.w, h1v.w, acc1);
        }
        gbuf[j] = acc0 + acc1;
        __syncthreads();
        if (j < M_HID) {
            float ii = gbuf[j], ff = gbuf[j + 128], gg = gbuf[j + 256], oo = gbuf[j + 384];
            float cn = sigmf(ff) * creg + sigmf(ii) * tanh_fast(gg);
            float hn = sigmf(oo) * tanh_fast(cn);
            creg = cn;
            hbuf[j] = hn;
            h1[((size_t)t * D_DIM + c) * M_HID + j] = (uint16_t)f2bf(hn);
        }
        __syncthreads();
    }
}

// ---------------------------------------------------------------------------
// Layer-1 grouped LSTM + Wl dot: 1 workgroup per channel, 512 threads.
// Wih1[c] and Whh1[c] both held in LDS as packed bf16 (2x128KB), f32 accum.
// Emits y[t,c] = relu(<h_new, Wl[c]> + bl[c]) via wave32 shfl reduction.
// ---------------------------------------------------------------------------
__global__ void lstm1_persistent(const float* __restrict__ Wih, const float* __restrict__ Whh,
                                 const float* __restrict__ bih, const float* __restrict__ bhh,
                                 const float* __restrict__ Wl,  const float* __restrict__ bl,
                                 const uint16_t* __restrict__ h1, float* __restrict__ y)
{
    extern __shared__ char smem[];
    uint4* WihP = (uint4*)smem;                 // [16][512] : 8 bf16 per uint4
    uint4* WhhP = WihP + 16 * G_DIM;            // [16][512]
    float* xs   = (float*)(WhhP + 16 * G_DIM);  // [128] layer-0 output (f32)
    float* hbuf = xs + M_HID;                   // [128]
    float* gbuf = hbuf + M_HID;                 // [512]
    float* ypart= gbuf + G_DIM;                 // [128]

    const int c = blockIdx.x;
    const int j = threadIdx.x;

    {   // convert both weight matrices to bf16 and stage in LDS (once)
        const float4* gi = (const float4*)(Wih + ((size_t)c * G_DIM + j) * M_HID);
        const float4* gh = (const float4*)(Whh + ((size_t)c * G_DIM + j) * M_HID);
#pragma unroll
        for (int m8 = 0; m8 < 16; ++m8) {
            float4 a = gi[m8 * 2], b = gi[m8 * 2 + 1];
            WihP[m8 * G_DIM + j] = make_uint4(pack2bf(a.x, a.y), pack2bf(a.z, a.w),
                                              pack2bf(b.x, b.y), pack2bf(b.z, b.w));
            a = gh[m8 * 2]; b = gh[m8 * 2 + 1];
            WhhP[m8 * G_DIM + j] = make_uint4(pack2bf(a.x, a.y), pack2bf(a.z, a.w),
                                              pack2bf(b.x, b.y), pack2bf(b.z, b.w));
        }
    }

    const float bsum = bih[(size_t)c * G_DIM + j] + bhh[(size_t)c * G_DIM + j];
    const float wl_j = (j < M_HID) ? Wl[(size_t)c * M_HID + j] : 0.0f;
    const float bl_c = bl[c];

    float creg = 0.0f;
    if (j < M_HID) hbuf[j] = 0.0f;
    __syncthreads();

    for (int t = 0; t < S_LEN; ++t) {
        if (j < M_HID)
            xs[j] = __uint_as_float((uint32_t)h1[((size_t)t * D_DIM + c) * M_HID + j] << 16);
        __syncthreads();

        // 4 accumulators: two independent 128-MAC dots, each split in half
        float a0 = bsum, a1 = 0.0f, a2 = 0.0f, a3 = 0.0f;
#pragma unroll
        for (int m8 = 0; m8 < 16; m8 += 2) {    // gi = Wih1 . x_t
            uint4 w0 = WihP[(m8 + 0) * G_DIM + j];
            uint4 w1 = WihP[(m8 + 1) * G_DIM + j];
            const float* xp = xs + m8 * 8;
            a0 = fmaf(bf_lo(w0.x), xp[0], a0);  a1 = fmaf(bf_hi(w0.x), xp[1], a1);
            a0 = fmaf(bf_lo(w0.y), xp[2], a0);  a1 = fmaf(bf_hi(w0.y), xp[3], a1);
            a0 = fmaf(bf_lo(w0.z), xp[4], a0);  a1 = fmaf(bf_hi(w0.z), xp[5], a1);
            a0 = fmaf(bf_lo(w0.w), xp[6], a0);  a1 = fmaf(bf_hi(w0.w), xp[7], a1);
            a0 = fmaf(bf_lo(w1.x), xp[8], a0);  a1 = fmaf(bf_hi(w1.x), xp[9], a1);
            a0 = fmaf(bf_lo(w1.y), xp[10], a0); a1 = fmaf(bf_hi(w1.y), xp[11], a1);
            a0 = fmaf(bf_lo(w1.z), xp[12], a0); a1 = fmaf(bf_hi(w1.z), xp[13], a1);
            a0 = fmaf(bf_lo(w1.w), xp[14], a0); a1 = fmaf(bf_hi(w1.w), xp[15], a1);
        }
#pragma unroll
        for (int m8 = 0; m8 < 16; m8 += 2) {    // + Whh1 . h_{t-1}
            uint4 w0 = WhhP[(m8 + 0) * G_DIM + j];
            uint4 w1 = WhhP[(m8 + 1) * G_DIM + j];
            const float* hp = hbuf + m8 * 8;
            a2 = fmaf(bf_lo(w0.x), hp[0], a2);  a3 = fmaf(bf_hi(w0.x), hp[1], a3);
            a2 = fmaf(bf_lo(w0.y), hp[2], a2);  a3 = fmaf(bf_hi(w0.y), hp[3], a3);
            a2 = fmaf(bf_lo(w0.z), hp[4], a2);  a3 = fmaf(bf_hi(w0.z), hp[5], a3);
            a2 = fmaf(bf_lo(w0.w), hp[6], a2);  a3 = fmaf(bf_hi(w0.w), hp[7], a3);
            a2 = fmaf(bf_lo(w1.x), hp[8], a2);  a3 = fmaf(bf_hi(w1.x), hp[9], a3);
            a2 = fmaf(bf_lo(w1.y), hp[10], a2); a3 = fmaf(bf_hi(w1.y), hp[11], a3);
            a2 = fmaf(bf_lo(w1.z), hp[12], a2); a3 = fmaf(bf_hi(w1.z), hp[13], a3);
            a2 = fmaf(bf_lo(w1.w), hp[14], a2); a3 = fmaf(bf_hi(w1.w), hp[15], a3);
        }
        gbuf[j] = (a0 + a1) + (a2 + a3);
        __syncthreads();
        if (j < M_HID) {
            float ii = gbuf[j], ff = gbuf[j + 128], gg = gbuf[j + 256], oo = gbuf[j + 384];
            float cn = sigmf(ff) * creg + sigmf(ii) * tanh_fast(gg);
            float hn = sigmf(oo) * tanh_fast(cn);
            creg = cn;
            hbuf[j] = hn;
            ypart[j] = hn * wl_j;
        }
        __syncthreads();
        if (j < 32) {                            // wave32 reduction of 128 partials
            float s = ypart[j] + ypart[j + 32] + ypart[j + 64] + ypart[j + 96];
#pragma unroll
            for (int off = 16; off > 0; off >>= 1)
                s += __shfl_down(s, off, 32);
            if (j == 0) y[t * D_DIM + c] = fmaxf(s + bl_c, 0.0f);
        }
        // ypart is only rewritten after the next gbuf barrier -> no extra barrier.
    }
}

// ---------------------------------------------------------------------------
extern "C" void kernel_launch(void* const* d_in, const int* in_sizes, int n_in,
                              void* d_out, int out_size, void* d_ws, size_t ws_size,
                              hipStream_t stream)
{
    const float* x     = (const float*)d_in[0];
    const float* W_in  = (const float*)d_in[1];
    const float* b_in  = (const float*)d_in[2];
    const float* Wih0  = (const float*)d_in[3];
    const float* Whh0  = (const float*)d_in[4];
    const float* bih0  = (const float*)d_in[5];
    const float* bhh0  = (const float*)d_in[6];
    const float* Wih1  = (const float*)d_in[7];
    const float* Whh1  = (const float*)d_in[8];
    const float* bih1  = (const float*)d_in[9];
    const float* bhh1  = (const float*)d_in[10];
    const float* Wl    = (const float*)d_in[11];
    const float* bl    = (const float*)d_in[12];
    const float* W_out = (const float*)d_in[13];
    const float* b_out = (const float*)d_in[14];
    float* out = (float*)d_out;

    char* ws = (char*)d_ws;
    float*    u  = (float*)ws;                                   // 4096*64 f32 (1MB)
    float*    yb = (float*)(ws + (size_t)S_LEN * D_DIM * 4);     // 4096*64 f32 (1MB)
    uint16_t* h1 = (uint16_t*)(ws + (size_t)S_LEN * D_DIM * 8);  // 4096*64*128 bf16 (64MB)

    // 1) u = relu(x @ W_in^T + b_in)            -- WMMA f32
    proj64_wmma<<<128, 256, 0, stream>>>(x, W_in, b_in, u, 0);

    // 2) layer-0 grouped LSTM (persistent, Whh0 in LDS f32)
    const size_t lds0 = (size_t)G_DIM * M_HID * 4 + (M_HID + G_DIM) * 4;
    lstm0_persistent<<<64, 512, lds0, stream>>>(Wih0, Whh0, bih0, bhh0, u, h1);

    // 3) layer-1 grouped LSTM + Wl reduction (persistent, both matrices bf16 in LDS)
    const size_t lds1 = (size_t)2 * 16 * G_DIM * 16 + (M_HID + M_HID + G_DIM + M_HID) * 4;
    lstm1_persistent<<<64, 512, lds1, stream>>>(Wih1, Whh1, bih1, bhh1, Wl, bl, h1, yb);

    // 4) out = sigmoid(y @ W_out^T + b_out)     -- WMMA f32
    proj64_wmma<<<128, 256, 0, stream>>>(yb, W_out, b_out, out, 1);
}